// CrossScaleVFE_24515673325995
// MI455X (gfx1250) — compile-verified
//
#include <hip/hip_runtime.h>

// ---------------------------------------------------------------------------
// CrossScaleVFE for MI455X (gfx1250, wave32).
// Per-pair KL is factored into per-fiber precomputes + an 80-long dot product,
// so the (n,m) cross stage is a 32x8x80 fp32 GEMM per grid cell done with
// V_WMMA_F32_16X16X4_F32. G/H intermediates are stored k-swizzled so each
// lane's fragment loads are contiguous b128s, and H is padded to 16 rows per
// cell (zero rows 8-15) so the WMMA loop is branch-free with EXEC all ones.
// ---------------------------------------------------------------------------

#define KDIM 8
#define NN 32
#define MM 8
#define MPAD 16                  // H rows per cell (8 real + 8 zero)
#define NCELL (48 * 48)          // GA*GB = 2304
#define GLEN 80                  // padded feature length (E 64, v 8, alpha, 1, pad 6)
#define NCHILD (NN * NCELL)      // 73728
#define NPARENT (MM * NCELL)     // 18432

typedef __attribute__((ext_vector_type(2))) float v2f;
typedef __attribute__((ext_vector_type(8))) float v8f;

// k-swizzle: logical feature k -> storage offset.
// k%4 in {0,1} -> [0..39] (lane-half 0), k%4 in {2,3} -> [40..79] (lane-half 1),
// so a lane's reads across the 20 WMMA k-steps are contiguous.
__host__ __device__ constexpr int swz(int k) {
  return (((k & 3) < 2) ? 0 : 40) + ((k >> 2) << 1) + (k & 1);
}

// ---- tiny fixed-size 8x8 linear algebra (fully unrolled, register resident) ----

__device__ __forceinline__ void lu8(float* A) {  // in-place LU, no pivoting
#pragma unroll
  for (int k = 0; k < 8; ++k) {
    float inv = 1.0f / A[k * 8 + k];
#pragma unroll
    for (int i = 0; i < 8; ++i) {
      if (i > k) {
        float f = A[i * 8 + k] * inv;
        A[i * 8 + k] = f;
#pragma unroll
        for (int j = 0; j < 8; ++j) {
          if (j > k) A[i * 8 + j] -= f * A[k * 8 + j];
        }
      }
    }
  }
}

__device__ __forceinline__ void lusolve8(const float* A, float* x) {
#pragma unroll
  for (int i = 1; i < 8; ++i) {  // forward, unit lower
    float s = x[i];
#pragma unroll
    for (int j = 0; j < 8; ++j) {
      if (j < i) s -= A[i * 8 + j] * x[j];
    }
    x[i] = s;
  }
#pragma unroll
  for (int i = 7; i >= 0; --i) {  // backward
    float s = x[i];
#pragma unroll
    for (int j = 0; j < 8; ++j) {
      if (j > i) s -= A[i * 8 + j] * x[j];
    }
    x[i] = s / A[i * 8 + i];
  }
}

__device__ __forceinline__ void chol8(float* A) {  // lower Cholesky in place (SPD)
#pragma unroll
  for (int j = 0; j < 8; ++j) {
    float d = A[j * 8 + j];
#pragma unroll
    for (int k = 0; k < 8; ++k) {
      if (k < j) d -= A[j * 8 + k] * A[j * 8 + k];
    }
    d = sqrtf(d);
    A[j * 8 + j] = d;
    float inv = 1.0f / d;
#pragma unroll
    for (int i = 0; i < 8; ++i) {
      if (i > j) {
        float s = A[i * 8 + j];
#pragma unroll
        for (int k = 0; k < 8; ++k) {
          if (k < j) s -= A[i * 8 + k] * A[j * 8 + k];
        }
        A[i * 8 + j] = s * inv;
      }
    }
  }
}

// ---------------------------------------------------------------------------
// Child stage: per fiber (n,a,b) build G = [E(64), v(8), alpha, 1, 0x6]
//   v = Oa^-1 mu,  C = Oa^-1 Sa Oa^-T,  E = C + v v^T,  alpha = -logdet(C)
// ---------------------------------------------------------------------------
__global__ void child_stage(const float* __restrict__ mu,
                            const float* __restrict__ sigma,
                            const float* __restrict__ omega,
                            float* __restrict__ G, int nf) {
  int f = blockIdx.x * blockDim.x + threadIdx.x;
  if (f >= nf) return;
  const float* Om = omega + (size_t)f * 64;
  const float* Sg = sigma + (size_t)f * 64;
  const float* Mu = mu + (size_t)f * 8;

  float A[64];
#pragma unroll
  for (int i = 0; i < 64; ++i) A[i] = Om[i];
  lu8(A);

  float v[8];
#pragma unroll
  for (int i = 0; i < 8; ++i) v[i] = Mu[i];
  lusolve8(A, v);

  // X = Oa^-1 * Sa  (column solves), stored into C
  float C[64];
#pragma unroll
  for (int j = 0; j < 8; ++j) {
    float x[8];
#pragma unroll
    for (int i = 0; i < 8; ++i) x[i] = Sg[i * 8 + j];
    lusolve8(A, x);
#pragma unroll
    for (int i = 0; i < 8; ++i) C[i * 8 + j] = x[i];
  }
  // C = X * Oa^-T : row j of C = Oa^-1 * (row j of X), in place row by row
#pragma unroll
  for (int j = 0; j < 8; ++j) {
    float x[8];
#pragma unroll
    for (int i = 0; i < 8; ++i) x[i] = C[j * 8 + i];
    lusolve8(A, x);
#pragma unroll
    for (int i = 0; i < 8; ++i) C[j * 8 + i] = x[i];
  }

  int n = f / NCELL, cell = f % NCELL;
  float* g = G + ((size_t)cell * NN + n) * GLEN;
#pragma unroll
  for (int i = 0; i < 8; ++i)
#pragma unroll
    for (int j = 0; j < 8; ++j) g[swz(i * 8 + j)] = C[i * 8 + j] + v[i] * v[j];
#pragma unroll
  for (int i = 0; i < 8; ++i) g[swz(64 + i)] = v[i];

  chol8(C);  // logdet C = 2 * sum log diag;  alpha = -logdet C
  float ld = 0.0f;
#pragma unroll
  for (int i = 0; i < 8; ++i) ld += logf(C[i * 8 + i]);
  g[swz(72)] = -2.0f * ld;
  g[swz(73)] = 1.0f;
#pragma unroll
  for (int i = 74; i < 80; ++i) g[swz(i)] = 0.0f;
}

// ---------------------------------------------------------------------------
// Parent stage: per fiber (m,a,b) build H = [P(64), -2q(8), 1, c+beta-8, 0x6]
//   Sb = L L^T, Z = L^-1 Ob, w = L^-1 mu, P = Z^T Z, q = Z^T w, c = w.w,
//   beta = -2 log|det Z|.  Also zeroes pad row m+8 of the 16-row cell block.
// ---------------------------------------------------------------------------
__global__ void parent_stage(const float* __restrict__ mu,
                             const float* __restrict__ sigma,
                             const float* __restrict__ omega,
                             float* __restrict__ H, int nf) {
  int f = blockIdx.x * blockDim.x + threadIdx.x;
  if (f >= nf) return;
  const float* Om = omega + (size_t)f * 64;
  const float* Sg = sigma + (size_t)f * 64;
  const float* Mu = mu + (size_t)f * 8;

  float L[64];
#pragma unroll
  for (int i = 0; i < 64; ++i) L[i] = Sg[i];
  chol8(L);

  float Z[64];
#pragma unroll
  for (int j = 0; j < 8; ++j) {
    float x[8];
#pragma unroll
    for (int i = 0; i < 8; ++i) x[i] = Om[i * 8 + j];
#pragma unroll
    for (int i = 0; i < 8; ++i) {  // forward solve, non-unit diag
      float s = x[i];
#pragma unroll
      for (int k = 0; k < 8; ++k) {
        if (k < i) s -= L[i * 8 + k] * x[k];
      }
      x[i] = s / L[i * 8 + i];
    }
#pragma unroll
    for (int i = 0; i < 8; ++i) Z[i * 8 + j] = x[i];
  }

  float w[8];
#pragma unroll
  for (int i = 0; i < 8; ++i) w[i] = Mu[i];
#pragma unroll
  for (int i = 0; i < 8; ++i) {
    float s = w[i];
#pragma unroll
    for (int k = 0; k < 8; ++k) {
      if (k < i) s -= L[i * 8 + k] * w[k];
    }
    w[i] = s / L[i * 8 + i];
  }

  int m = f / NCELL, cell = f % NCELL;
  float* h = H + ((size_t)cell * MPAD + m) * GLEN;
#pragma unroll
  for (int i = 0; i < 8; ++i)
#pragma unroll
    for (int j = 0; j < 8; ++j) {
      float s = 0.0f;
#pragma unroll
      for (int k = 0; k < 8; ++k) s += Z[k * 8 + i] * Z[k * 8 + j];
      h[swz(i * 8 + j)] = s;  // P = Z^T Z
    }
  float c = 0.0f;
#pragma unroll
  for (int i = 0; i < 8; ++i) {
    float s = 0.0f;
#pragma unroll
    for (int k = 0; k < 8; ++k) s += Z[k * 8 + i] * w[k];
    h[swz(64 + i)] = -2.0f * s;  // -2 q
    c += w[i] * w[i];
  }

  lu8(Z);  // log|det Z|
  float ld = 0.0f;
#pragma unroll
  for (int i = 0; i < 8; ++i) ld += logf(fabsf(Z[i * 8 + i]));
  h[swz(72)] = 1.0f;
  h[swz(73)] = c - 2.0f * ld - 8.0f;  // c + beta - K
#pragma unroll
  for (int i = 74; i < 80; ++i) h[swz(i)] = 0.0f;

  // zero pad row (m+8) so the WMMA B-fragment loads are branch-free
  float* hp = H + ((size_t)cell * MPAD + 8 + m) * GLEN;
#pragma unroll
  for (int i = 0; i < 80; i += 4) {
    float4 z4 = {0.f, 0.f, 0.f, 0.f};
    *(float4*)(hp + i) = z4;
  }
}

// ---------------------------------------------------------------------------
// Pair stage: per (cell, term) compute S = G(32x80) * H^T(80x8pad16) with
// V_WMMA_F32_16X16X4_F32 (wave32), then the W-weighted sum over (n,m).
// 2 waves per block: wave w owns rows n0 = 16w .. 16w+15. Thanks to the
// k-swizzle each b128 load feeds two consecutive WMMA k-steps, branch-free.
// ---------------------------------------------------------------------------
__global__ void gemm_reduce(const float* __restrict__ G0,
                            const float* __restrict__ G1,
                            const float* __restrict__ H0,
                            const float* __restrict__ H1,
                            const float* __restrict__ W,
                            float* __restrict__ partial) {
  int cell = blockIdx.x;
  int term = blockIdx.y;
  const float* G = (term == 0 ? G0 : G1) + (size_t)cell * NN * GLEN;
  const float* H = (term == 0 ? H0 : H1) + (size_t)cell * MPAD * GLEN;

  int lane = threadIdx.x & 31;
  int wave = threadIdx.x >> 5;
  int n0 = wave * 16;

  int row = n0 + (lane & 15);  // A-matrix row for this lane
  int col = lane & 15;         // B-matrix column for this lane
  int half = lane >> 4;        // lanes 16-31 hold K = k0+2, k0+3

  const float* Grow = G + (size_t)row * GLEN + 40 * half;
  const float* Hrow = H + (size_t)col * GLEN + 40 * half;

  v8f acc = {0.f, 0.f, 0.f, 0.f, 0.f, 0.f, 0.f, 0.f};
#pragma unroll
  for (int t = 0; t < 10; ++t) {  // each iteration covers two WMMA k-steps
    float4 av = *(const float4*)(Grow + 4 * t);
    float4 bv = *(const float4*)(Hrow + 4 * t);
    v2f a0 = {av.x, av.y};
    v2f b0 = {bv.x, bv.y};
    // 8 args: (neg_a, A, neg_b, B, c_mod, C, reuse_a, reuse_b)
    acc = __builtin_amdgcn_wmma_f32_16x16x4_f32(
        false, a0, false, b0, (short)0, acc, false, false);
    v2f a1 = {av.z, av.w};
    v2f b1 = {bv.z, bv.w};
    acc = __builtin_amdgcn_wmma_f32_16x16x4_f32(
        false, a1, false, b1, (short)0, acc, false, false);
  }

  // acc[r] = S[n0 + r + 8*(lane>=16), lane&15]  (C/D layout, 32-bit 16x16)
  float p = 0.0f;
  if (col < MM) {
    int hi = half * 8;
#pragma unroll
    for (int r = 0; r < 8; ++r) {
      int n = n0 + hi + r;
      p += W[(size_t)(n * MM + col) * NCELL + cell] * acc[r];
    }
  }
#pragma unroll
  for (int off = 16; off > 0; off >>= 1) p += __shfl_down(p, off, 32);

  __shared__ float sred[2];
  if (lane == 0) sred[wave] = p;
  __syncthreads();
  if (threadIdx.x == 0)
    partial[(size_t)term * NCELL + cell] = 0.5f * (sred[0] + sred[1]);
}

// ---------------------------------------------------------------------------
// Deterministic fixed-order final reduction of per-cell partials.
// ---------------------------------------------------------------------------
__global__ void final_reduce(const float* __restrict__ partial,
                             float* __restrict__ out) {
  __shared__ float sa[256];
  __shared__ float sb[256];
  int tid = threadIdx.x;
  float s0 = 0.f, s1 = 0.f;
  for (int i = tid; i < NCELL; i += 256) {
    s0 += partial[i];
    s1 += partial[NCELL + i];
  }
  sa[tid] = s0;
  sb[tid] = s1;
  __syncthreads();
  for (int off = 128; off > 0; off >>= 1) {
    if (tid < off) {
      sa[tid] += sa[tid + off];
      sb[tid] += sb[tid + off];
    }
    __syncthreads();
  }
  if (tid == 0) {
    float belief = sa[0], model = sb[0];
    out[0] = belief + model;  // LAMBDA_BELIEF = LAMBDA_MODEL = 1
    out[1] = belief;
    out[2] = model;
  }
}

// ---------------------------------------------------------------------------

extern "C" void kernel_launch(void* const* d_in, const int* in_sizes, int n_in,
                              void* d_out, int out_size, void* d_ws,
                              size_t ws_size, hipStream_t stream) {
  (void)in_sizes; (void)n_in; (void)out_size; (void)ws_size;
  const float* mu_p          = (const float*)d_in[0];
  const float* sigma_p       = (const float*)d_in[1];
  const float* omega_c       = (const float*)d_in[2];
  const float* mu_q          = (const float*)d_in[3];
  const float* sigma_q       = (const float*)d_in[4];
  const float* omega_par     = (const float*)d_in[5];
  const float* mu_r          = (const float*)d_in[6];
  const float* sigma_r       = (const float*)d_in[7];
  const float* omega_model_c = (const float*)d_in[8];
  const float* mu_s          = (const float*)d_in[9];
  const float* sigma_s       = (const float*)d_in[10];
  const float* omega_model_p = (const float*)d_in[11];
  const float* W             = (const float*)d_in[12];

  float* G0 = (float*)d_ws;
  size_t gsz = (size_t)NCHILD * GLEN;          // 5,898,240 floats
  size_t hsz = (size_t)NCELL * MPAD * GLEN;    // 2,949,120 floats (padded)
  float* G1 = G0 + gsz;
  float* H0 = G1 + gsz;
  float* H1 = H0 + hsz;
  float* partial = H1 + hsz;                   // 2 * NCELL floats

  child_stage<<<NCHILD / 128, 128, 0, stream>>>(mu_p, sigma_p, omega_c, G0, NCHILD);
  child_stage<<<NCHILD / 128, 128, 0, stream>>>(mu_r, sigma_r, omega_model_c, G1, NCHILD);
  parent_stage<<<NPARENT / 128, 128, 0, stream>>>(mu_q, sigma_q, omega_par, H0, NPARENT);
  parent_stage<<<NPARENT / 128, 128, 0, stream>>>(mu_s, sigma_s, omega_model_p, H1, NPARENT);

  dim3 grid(NCELL, 2);
  gemm_reduce<<<grid, 64, 0, stream>>>(G0, G1, H0, H1, W, partial);
  final_reduce<<<1, 256, 0, stream>>>(partial, (float*)d_out);
}